// QuantumEntangledNeuron_3573412790535
// MI455X (gfx1250) — compile-verified
//
#include <hip/hip_runtime.h>
#include <math.h>

// ---------------------------------------------------------------------------
// QuantumEntangledNeuron forward for gfx1250 (MI455X), fp32 WMMA everywhere.
// B=64, T=128, I=4096, Q=256, W=32, THR=0.2
// Round 3: fix async-to-LDS builtin signature (v4i* in AS1/AS3).
// ---------------------------------------------------------------------------

typedef __attribute__((ext_vector_type(2))) float v2f;
typedef __attribute__((ext_vector_type(8))) float v8f;
typedef int v4i __attribute__((vector_size(16)));

#if __has_builtin(__builtin_amdgcn_global_load_async_to_lds_b128)
#define HAVE_ASYNC_LDS 1
#else
#define HAVE_ASYNC_LDS 0
#endif

namespace qen {
constexpr int Bn = 64, Tn = 128, In = 4096, Qn = 256, Wn = 32;
constexpr float THR = 0.2f;

// workspace offsets (float elements)
constexpr size_t OFF_AR  = 0;                                  // [B*T, Q]
constexpr size_t OFF_AI  = OFF_AR  + (size_t)Bn * Tn * Qn;     // [B*T, Q]
constexpr size_t OFF_PHI = OFF_AI  + (size_t)Bn * Tn * Qn;     // [B*T]
constexpr size_t OFF_CM  = OFF_PHI + (size_t)Bn * Tn;          // [Q, Q]
constexpr size_t OFF_CH  = OFF_CM  + (size_t)Qn * Qn;          // [W, Q]
constexpr size_t OFF_CI  = OFF_CH  + (size_t)Wn * Qn;          // [B, Q]
constexpr size_t OFF_ESR = OFF_CI  + (size_t)Bn * Qn;          // [B, Q]
constexpr size_t OFF_ESI = OFF_ESR + (size_t)Bn * Qn;          // [B, Q]
constexpr size_t OFF_SP  = OFF_ESI + (size_t)Bn * Qn;          // [B, Q]
constexpr size_t OFF_CL  = OFF_SP  + (size_t)Bn * Qn;          // scalar
constexpr size_t OFF_MA  = OFF_CL + 1;                         // scalar

// d_out layout (float elements)
constexpr size_t OUT_OUTPUTS = 0;                              // [B, T, Q]
constexpr size_t OUT_TRACE   = (size_t)Bn * Tn * Qn;           // [B, T, Q]
constexpr size_t OUT_ENT     = OUT_TRACE + (size_t)Bn * Tn * Qn; // [Q, T]
constexpr size_t OUT_SCAL    = OUT_ENT + (size_t)Qn * Tn;      // ch.mean, cl, ma

// LDS tile geometry for the big GEMM
constexpr int KC = 32;        // K-chunk
constexpr int LSTR = 36;      // padded row stride (floats): 144B keeps 16B
                              // alignment for b128 and spreads A lanes over
                              // 16 distinct banks (36*l mod 64 distinct)
} // namespace qen
using namespace qen;

__device__ __forceinline__ v8f wmma_f32(v2f a, v2f b, v8f c) {
    // V_WMMA_F32_16X16X4_F32: (neg_a, A, neg_b, B, c_mod, C, reuse_a, reuse_b)
    return __builtin_amdgcn_wmma_f32_16x16x4_f32(false, a, false, b,
                                                 (short)0, c, false, false);
}

__device__ __forceinline__ float sigm(float x) { return 1.0f / (1.0f + expf(-x)); }

// 16B global -> LDS copy: async DMA path on CDNA5, VGPR round-trip fallback.
__device__ __forceinline__ void g2lds_b128(const float* g, float* l) {
#if HAVE_ASYNC_LDS
    __builtin_amdgcn_global_load_async_to_lds_b128(
        (__attribute__((address_space(1))) v4i*)g,
        (__attribute__((address_space(3))) v4i*)l, 0, 0);
#else
    *(float4*)l = *(const float4*)g;
#endif
}

__device__ __forceinline__ void wait_g2lds() {
#if HAVE_ASYNC_LDS
#if __has_builtin(__builtin_amdgcn_s_wait_asynccnt)
    __builtin_amdgcn_s_wait_asynccnt(0);
#else
    asm volatile("s_wait_asynccnt 0x0" ::: "memory");
#endif
#endif
}

// ---------------------------------------------------------------------------
// Init: cm workspace copy, ch=0, cl=0, ma=ma_in. Grid: 256x256.
// ---------------------------------------------------------------------------
__global__ void k_init(const float* __restrict__ cm_in,
                       const float* __restrict__ ma_in,
                       float* __restrict__ ws) {
    int i = blockIdx.x * blockDim.x + threadIdx.x;
    if (i < Qn * Qn) ws[OFF_CM + i] = cm_in[i];
    if (i < Wn * Qn) ws[OFF_CH + i] = 0.0f;
    if (i == 0) { ws[OFF_CL] = 0.0f; ws[OFF_MA] = ma_in[0]; }
}

// ---------------------------------------------------------------------------
// AR = X @ Qw_real, AI = X @ Qw_imag   (M=8192, K=4096, N=256)
// Block = 256 threads = 8 waves arranged 4x2, computing a 64x32 output tile.
// Per K-chunk (KC=32): stage X 64x32 and Qr/Qi 32x32 tiles into LDS with
// coalesced 128-bit async copies, then sweep WMMA from LDS.
// Grid: (8192/64) x (256/32) = 128 x 8 = 1024 blocks.
// ---------------------------------------------------------------------------
__global__ void __launch_bounds__(256) k_gemm_xw(const float* __restrict__ X,
                                                 const float* __restrict__ Qr,
                                                 const float* __restrict__ Qi,
                                                 float* __restrict__ ws) {
    float* AR = ws + OFF_AR;
    float* AI = ws + OFF_AI;

    __shared__ float Xs[64 * LSTR];   // [row][k], padded
    __shared__ float Brs[KC * LSTR];  // [k][n],  padded
    __shared__ float Bis[KC * LSTR];

    const int tid  = threadIdx.x;
    const int lane = tid & 31, half = lane >> 4, l16 = lane & 15;
    const int wave = tid >> 5;            // 0..7
    const int wm = wave >> 1;             // 0..3 (16-row tile inside 64)
    const int wn = wave & 1;              // 0..1 (16-col tile inside 32)
    const int bm = blockIdx.x >> 3;       // 0..127
    const int bn = blockIdx.x & 7;        // 0..7
    const int row0 = bm * 64;             // global M base
    const int col0 = bn * 32;             // global N base

    // staging assignments (float4 granularity)
    // X tile: 64 rows x 8 float4 -> 512 xfers -> 2 per thread
    const int xr0 = tid >> 3;             // f4 idx 0..255 -> row 0..31
    const int xc4 = (tid & 7) * 4;        // float offset in row (0,4,..28)
    // B tiles: 32 rows x 8 float4 -> 256 xfers -> 1 per thread
    const int br_ = tid >> 3;             // k row 0..31
    const int bc4 = (tid & 7) * 4;

    v8f accR = {}; v8f accI = {};

    for (int k0 = 0; k0 < In; k0 += KC) {
        // ---- stage tiles into LDS (coalesced 128-bit, async on CDNA5) ----
        g2lds_b128(&X[(size_t)(row0 + xr0) * In + k0 + xc4],
                   &Xs[xr0 * LSTR + xc4]);
        g2lds_b128(&X[(size_t)(row0 + xr0 + 32) * In + k0 + xc4],
                   &Xs[(xr0 + 32) * LSTR + xc4]);
        g2lds_b128(&Qr[(size_t)(k0 + br_) * Qn + col0 + bc4],
                   &Brs[br_ * LSTR + bc4]);
        g2lds_b128(&Qi[(size_t)(k0 + br_) * Qn + col0 + bc4],
                   &Bis[br_ * LSTR + bc4]);
        // prefetch next chunk of X into cache while this chunk computes
        if (k0 + KC < In)
            __builtin_prefetch(&X[(size_t)(row0 + xr0) * In + k0 + KC + xc4], 0, 1);
        wait_g2lds();
        __syncthreads();

        // ---- WMMA sweep over the chunk ----
        const float* xrow = &Xs[(wm * 16 + l16) * LSTR];
        for (int kk = 0; kk < KC; kk += 4) {
            const int k = kk + 2 * half;
            v2f a;  a[0]  = xrow[k];                 a[1]  = xrow[k + 1];
            v2f br; br[0] = Brs[k * LSTR + wn * 16 + l16];
                    br[1] = Brs[(k + 1) * LSTR + wn * 16 + l16];
            v2f bi; bi[0] = Bis[k * LSTR + wn * 16 + l16];
                    bi[1] = Bis[(k + 1) * LSTR + wn * 16 + l16];
            accR = wmma_f32(a, br, accR);
            accI = wmma_f32(a, bi, accI);
        }
        __syncthreads();
    }

    const int col = col0 + wn * 16 + l16;
    for (int r = 0; r < 8; ++r) {
        const int m = row0 + wm * 16 + r + 8 * half;
        AR[(size_t)m * Qn + col] = accR[r];
        AI[(size_t)m * Qn + col] = accI[r];
    }
}

// ---------------------------------------------------------------------------
// phi[row] for all 8192 rows of AR. One 128-thread block per row.
// H(p1)+H(p2)-H(joint), 10/10/100 bins, natural log, RNE rounding.
// ---------------------------------------------------------------------------
__global__ void __launch_bounds__(128) k_phi(float* __restrict__ ws) {
    const float* AR = ws + OFF_AR;
    float* PHI = ws + OFF_PHI;
    __shared__ int h1[10], h2[10], hj[100];
    __shared__ float red[128];
    const int t = threadIdx.x;
    const int row = blockIdx.x;
    if (t < 10) { h1[t] = 0; h2[t] = 0; }
    if (t < 100) hj[t] = 0;
    __syncthreads();
    int d1 = (int)rintf(AR[(size_t)row * Qn + t] * 10.0f);
    int d2 = (int)rintf(AR[(size_t)row * Qn + 128 + t] * 10.0f);
    d1 = d1 < 0 ? 0 : (d1 > 9 ? 9 : d1);
    d2 = d2 < 0 ? 0 : (d2 > 9 ? 9 : d2);
    atomicAdd(&h1[d1], 1);
    atomicAdd(&h2[d2], 1);
    atomicAdd(&hj[d1 * 10 + d2], 1);
    __syncthreads();
    auto ent = [](int c) -> float {
        if (c <= 0) return 0.0f;
        float p = (float)c * (1.0f / 128.0f);
        return -p * logf(p);
    };
    float local = 0.0f;
    if (t < 10)  local += ent(h1[t]) + ent(h2[t]);
    if (t < 100) local -= ent(hj[t]);
    red[t] = local;
    __syncthreads();
    for (int s = 64; s > 0; s >>= 1) {
        if (t < s) red[t] += red[t + s];
        __syncthreads();
    }
    if (t == 0) PHI[row] = fmaxf(0.0f, red[0]);
}

// ---------------------------------------------------------------------------
// Step front: mm = sigmoid((AR_t @ cm) * ma); ci = mm * gate*phi;
//             es = (AR_t@E + i AI_t@E) * ci
// Three WMMA accumulators share A fragments. 64x256 output = 64 tiles.
// Grid: 8 blocks x 256 threads.
// ---------------------------------------------------------------------------
__global__ void __launch_bounds__(256) k_step_front(const float* __restrict__ E,
                                                    float* __restrict__ ws, int t) {
    const float* AR  = ws + OFF_AR;
    const float* AI  = ws + OFF_AI;
    const float* CM  = ws + OFF_CM;
    const float* PHI = ws + OFF_PHI;
    float* CI  = ws + OFF_CI;
    float* ESR = ws + OFF_ESR;
    float* ESI = ws + OFF_ESI;
    const float ma = ws[OFF_MA];
    const int lane = threadIdx.x & 31, half = lane >> 4, l16 = lane & 15;
    const int wave = blockIdx.x * (blockDim.x >> 5) + (threadIdx.x >> 5); // 0..63
    const int tm = wave >> 4;          // 0..3 (batch tiles)
    const int tn = wave & 15;          // 0..15
    const int b = tm * 16 + l16;
    const int col = tn * 16 + l16;
    const size_t rowA = ((size_t)b * Tn + t) * Qn;
    v8f amm = {}, aer = {}, aei = {};
    for (int k0 = 0; k0 < Qn; k0 += 4) {
        const int k = k0 + 2 * half;
        v2f aR; aR[0] = AR[rowA + k];              aR[1] = AR[rowA + k + 1];
        v2f aIv; aIv[0] = AI[rowA + k];            aIv[1] = AI[rowA + k + 1];
        v2f bc; bc[0] = CM[(size_t)k * Qn + col];  bc[1] = CM[(size_t)(k + 1) * Qn + col];
        v2f be; be[0] = E[(size_t)k * Qn + col];   be[1] = E[(size_t)(k + 1) * Qn + col];
        amm = wmma_f32(aR, bc, amm);
        aer = wmma_f32(aR, be, aer);
        aei = wmma_f32(aIv, be, aei);
    }
    for (int r = 0; r < 8; ++r) {
        const int bb = tm * 16 + r + 8 * half;
        const float phiv = PHI[(size_t)bb * Tn + t];
        const float mmv = sigm(amm[r] * ma);
        const float g = (phiv > THR) ? phiv : 0.0f;
        const float civ = mmv * g;
        const size_t idx = (size_t)bb * Qn + col;
        CI[idx]  = civ;
        ESR[idx] = aer[r] * civ;
        ESI[idx] = aei[r] * civ;
    }
}

// ---------------------------------------------------------------------------
// Step middle: sequential pair-entangle, ch/cohm, spikes/trace/sp, cl, ma.
// Single block of 256 threads (thread q owns column q of all [B,Q] arrays).
// ---------------------------------------------------------------------------
__global__ void __launch_bounds__(256) k_step_mid(const float* __restrict__ phase,
                                                  float* __restrict__ out,
                                                  float* __restrict__ ws, int t) {
    float* CI  = ws + OFF_CI;
    float* ESR = ws + OFF_ESR;
    float* ESI = ws + OFF_ESI;
    float* CH  = ws + OFF_CH;
    float* SP  = ws + OFF_SP;
    const float* PHI = ws + OFF_PHI;
    __shared__ float red[256];
    const int q = threadIdx.x;

    // --- sequential conditional pair mixing: k = 0 .. 2*(B-1)-1 ---
    for (int k = 0; k < 2 * (Bn - 1); ++k) {
        const int i = k >> 1;
        const int jr = i + 1 + (k & 1);
        const int j = (jr < Bn) ? jr : (Bn - 1);
        const bool valid = jr < Bn;
        red[q] = CI[(size_t)i * Qn + q] * CI[(size_t)j * Qn + q];
        __syncthreads();
        for (int s = 128; s > 0; s >>= 1) {
            if (q < s) red[q] += red[q + s];
            __syncthreads();
        }
        const float c = red[0];
        __syncthreads();
        if ((c > 0.6f) && valid) {
            const size_t ii = (size_t)i * Qn + q, jj = (size_t)j * Qn + q;
            const float eir = 0.7f * ESR[ii] + 0.3f * ESR[jj];
            const float eii = 0.7f * ESI[ii] + 0.3f * ESI[jj];
            const float ejr = 0.7f * ESR[jj] + 0.3f * eir;
            const float eji = 0.7f * ESI[jj] + 0.3f * eii;
            ESR[ii] = eir; ESI[ii] = eii;
            ESR[jj] = ejr; ESI[jj] = eji;
        }
    }

    // --- ch[t%W][q] = mean_b |es| ; cohm = exp(-var_ddof1(ch[:,q])) ---
    {
        float s = 0.0f;
        for (int b = 0; b < Bn; ++b) {
            const size_t idx = (size_t)b * Qn + q;
            s += sqrtf(ESR[idx] * ESR[idx] + ESI[idx] * ESI[idx]);
        }
        CH[(size_t)(t % Wn) * Qn + q] = s * (1.0f / Bn);
        float mean = 0.0f;
        for (int w = 0; w < Wn; ++w) mean += CH[(size_t)w * Qn + q];
        mean *= (1.0f / Wn);
        float var = 0.0f;
        for (int w = 0; w < Wn; ++w) {
            const float d = CH[(size_t)w * Qn + q] - mean;
            var += d * d;
        }
        var *= (1.0f / (Wn - 1));
        out[OUT_ENT + (size_t)q * Tn + t] = expf(-var);
    }

    // --- sp / spikes / outputs / trace ---
    float ci_col_sum = 0.0f;
    for (int b = 0; b < Bn; ++b) {
        const size_t idx = (size_t)b * Qn + q;
        const float civ = CI[idx];
        ci_col_sum += civ;
        const float m2 = ESR[idx] * ESR[idx] + ESI[idx] * ESI[idx];
        const float spv = m2 * sigm(2.0f * civ);
        const size_t gidx = ((size_t)b * Tn + t) * Qn + q;
        const float spike = (spv > phase[gidx]) ? 1.0f : 0.0f;
        out[OUT_OUTPUTS + gidx] = (spike + spv) - spv;  // straight-through fwd
        out[OUT_TRACE + gidx]   = civ;
        SP[idx] = spv;
    }

    // --- cl = 0.9*cl + 0.1*mean_b(phi_t) ---
    red[q] = (q < Bn) ? PHI[(size_t)q * Tn + t] : 0.0f;
    __syncthreads();
    for (int s = 128; s > 0; s >>= 1) {
        if (q < s) red[q] += red[q + s];
        __syncthreads();
    }
    if (q == 0) ws[OFF_CL] = 0.9f * ws[OFF_CL] + 0.1f * (red[0] * (1.0f / Bn));
    __syncthreads();

    // --- ma += (mean(ci) - ma) * 0.01 ---
    red[q] = ci_col_sum;
    __syncthreads();
    for (int s = 128; s > 0; s >>= 1) {
        if (q < s) red[q] += red[q + s];
        __syncthreads();
    }
    if (q == 0) {
        const float m = red[0] * (1.0f / (Bn * Qn));
        ws[OFF_MA] += (m - ws[OFF_MA]) * 0.01f;
    }
}

// ---------------------------------------------------------------------------
// Step cm-update: ac = sp^T @ ci (256x256, K=64); cm += ac * cl * 1e-4.
// 256 tiles of 16x16; grid: 32 blocks x 256 threads = 256 waves.
// ---------------------------------------------------------------------------
__global__ void __launch_bounds__(256) k_step_cm(float* __restrict__ ws) {
    float* CM = ws + OFF_CM;
    const float* SP = ws + OFF_SP;
    const float* CI = ws + OFF_CI;
    const float scale = ws[OFF_CL] * 0.1f * 0.001f;
    const int lane = threadIdx.x & 31, half = lane >> 4, l16 = lane & 15;
    const int wave = blockIdx.x * (blockDim.x >> 5) + (threadIdx.x >> 5); // 0..255
    const int tm = wave >> 4;
    const int tn = wave & 15;
    const int rowm = tm * 16 + l16;  // m index into A = sp^T
    const int col = tn * 16 + l16;
    v8f acc = {};
    for (int k0 = 0; k0 < Bn; k0 += 4) {
        const int k = k0 + 2 * half;
        v2f a; a[0] = SP[(size_t)k * Qn + rowm]; a[1] = SP[(size_t)(k + 1) * Qn + rowm];
        v2f b; b[0] = CI[(size_t)k * Qn + col];  b[1] = CI[(size_t)(k + 1) * Qn + col];
        acc = wmma_f32(a, b, acc);
    }
    for (int r = 0; r < 8; ++r) {
        const int m = tm * 16 + r + 8 * half;
        CM[(size_t)m * Qn + col] += acc[r] * scale;
    }
}

// ---------------------------------------------------------------------------
// Final scalars: ch.mean(), cl, ma.
// ---------------------------------------------------------------------------
__global__ void __launch_bounds__(256) k_final(float* __restrict__ out,
                                               const float* __restrict__ ws) {
    __shared__ float red[256];
    float s = 0.0f;
    for (int i = threadIdx.x; i < Wn * Qn; i += 256) s += ws[OFF_CH + i];
    red[threadIdx.x] = s;
    __syncthreads();
    for (int st = 128; st > 0; st >>= 1) {
        if (threadIdx.x < st) red[threadIdx.x] += red[threadIdx.x + st];
        __syncthreads();
    }
    if (threadIdx.x == 0) {
        out[OUT_SCAL + 0] = red[0] * (1.0f / (Wn * Qn));
        out[OUT_SCAL + 1] = ws[OFF_CL];
        out[OUT_SCAL + 2] = ws[OFF_MA];
    }
}

// ---------------------------------------------------------------------------
extern "C" void kernel_launch(void* const* d_in, const int* in_sizes, int n_in,
                              void* d_out, int out_size, void* d_ws, size_t ws_size,
                              hipStream_t stream) {
    (void)in_sizes; (void)n_in; (void)out_size; (void)ws_size;
    const float* x     = (const float*)d_in[0];  // (B,T,I)
    const float* phase = (const float*)d_in[1];  // (B,T,Q)
    const float* qr    = (const float*)d_in[2];  // (I,Q)
    const float* qi    = (const float*)d_in[3];  // (I,Q)
    const float* cm    = (const float*)d_in[4];  // (Q,Q)
    const float* ma    = (const float*)d_in[5];  // scalar
    const float* E     = (const float*)d_in[6];  // (Q,Q)
    float* out = (float*)d_out;
    float* ws  = (float*)d_ws;

    k_init<<<256, 256, 0, stream>>>(cm, ma, ws);
    k_gemm_xw<<<1024, 256, 0, stream>>>(x, qr, qi, ws);
    k_phi<<<Bn * Tn, 128, 0, stream>>>(ws);
    for (int t = 0; t < Tn; ++t) {
        k_step_front<<<8, 256, 0, stream>>>(E, ws, t);
        k_step_mid<<<1, 256, 0, stream>>>(phase, out, ws, t);
        k_step_cm<<<32, 256, 0, stream>>>(ws);
    }
    k_final<<<1, 256, 0, stream>>>(out, ws);
}